// SUB_IO_Loss_44796508897295
// MI455X (gfx1250) — compile-verified
//
#include <hip/hip_runtime.h>
#include <hip/hip_bf16.h>

// Problem constants from the reference
#define B_SZ 512
#define D_SZ 256
#define M_SZ 256
#define INV_TEMP (1.0f / 0.07f)

typedef __attribute__((ext_vector_type(2))) float v2f;
typedef __attribute__((ext_vector_type(8))) float v8f;

// ---------------------------------------------------------------------------
// Kernel 1: sim[m, n] = dot(features_o[idx[m]], features_i[n]) * (1/TEMP)
// One wave32 per 16x16 output tile, V_WMMA_F32_16X16X4_F32 over K = D in
// steps of 4. f32 WMMA keeps the reference's f32 GEMM semantics exactly.
//
// ISA layouts (CDNA5 ISA):
//   A 16x4 f32 : lanes 0-15 hold (M=lane, K=k..k+1) in VGPR0/1,
//                lanes 16-31 hold (M=lane-16, K=k+2..k+3)
//   B 4x16 f32 : symmetric with N in lanes (B(k,n) = features_i[n,k],
//                so each lane loads a contiguous float2 from row n)
//   C/D 16x16  : VGPR r -> M=r (lanes 0-15) / M=r+8 (lanes 16-31), N=lane%16
// ---------------------------------------------------------------------------
__global__ __launch_bounds__(32)
void sim_wmma_kernel(const float* __restrict__ fo,
                     const float* __restrict__ fi,
                     const int*   __restrict__ idx,
                     float*       __restrict__ sim)
{
    const int ntile = blockIdx.x;        // 0..31 (B/16)
    const int mtile = blockIdx.y;        // 0..15 (M/16)
    const int lane  = threadIdx.x;       // 0..31
    const int half  = lane >> 4;         // 0: K pair 0, 1: K pair 1
    const int lmod  = lane & 15;

    const int  m    = mtile * 16 + lmod;
    const int  n    = ntile * 16 + lmod;
    const long rowA = (long)idx[m] * D_SZ;   // gathered anchor row
    const long rowB = (long)n      * D_SZ;

    v8f c = {};
#pragma unroll 4
    for (int k = 0; k < D_SZ; k += 4) {
        const int kk = k + half * 2;
        v2f a = *(const v2f*)(fo + rowA + kk);
        v2f b = *(const v2f*)(fi + rowB + kk);
        // (neg_a, A, neg_b, B, c_mod, C, reuse_a, reuse_b)
        c = __builtin_amdgcn_wmma_f32_16x16x4_f32(
                false, a, false, b, (short)0, c, false, false);
    }

    const int row0 = mtile * 16 + half * 8;
    float* out = sim + (long)row0 * B_SZ + ntile * 16 + lmod;
#pragma unroll
    for (int r = 0; r < 8; ++r)
        out[(long)r * B_SZ] = c[r] * INV_TEMP;
}

// ---------------------------------------------------------------------------
// Kernel 2: per row m, compact pos/neg sim values into LDS with a
// deterministic wave32-ballot prefix (no LDS atomics -> bitwise reproducible
// across graph replays), then accumulate
//   sum_{p in pos} sum_{n in neg} softplus(sim_n - sim_p)
// softplus(x) = max(x,0) + log(1 + exp(-|x|))  (== -log_sigmoid(sp - sn))
// One 512-thread block per row (one thread per column).
// ---------------------------------------------------------------------------
__global__ __launch_bounds__(512)
void row_loss_kernel(const float* __restrict__ sim,
                     const float* __restrict__ mask,
                     const int*   __restrict__ idx,
                     float*       __restrict__ row_loss,
                     float*       __restrict__ row_pair)
{
    __shared__ float s_pos[B_SZ];
    __shared__ float s_neg[B_SZ];
    __shared__ int   s_wcnt[16];         // pos count per wave (16 waves)
    __shared__ float s_red[512];

    const int m   = blockIdx.x;
    const int tid = threadIdx.x;         // == column j
    const int wid = tid >> 5;
    const int lid = tid & 31;

    const float sv = sim [(long)m      * B_SZ + tid];
    const float mv = mask[(long)idx[m] * B_SZ + tid];
    const bool  isPos = (mv > 0.5f);     // mask is exactly 0.0 / 1.0

    const unsigned ball   = (unsigned)__ballot(isPos);   // wave32: low 32 bits
    const int      popcLt = __popc(ball & ((1u << lid) - 1u));
    if (lid == 0) s_wcnt[wid] = __popc(ball);
    __syncthreads();

    int posBefore = 0, totPos = 0;
#pragma unroll
    for (int w = 0; w < 16; ++w) {
        const int cw = s_wcnt[w];
        if (w < wid) posBefore += cw;
        totPos += cw;
    }
    if (isPos) s_pos[posBefore + popcLt] = sv;
    else       s_neg[(wid * 32 - posBefore) + (lid - popcLt)] = sv;
    __syncthreads();

    const int npos = totPos;
    const int nneg = B_SZ - totPos;

    float acc = 0.0f;
    for (int pi = 0; pi < npos; ++pi) {
        const float sp = s_pos[pi];
        for (int ni = tid; ni < nneg; ni += 512) {
            const float x  = s_neg[ni] - sp;        // sim_n - sim_p
            const float ax = fabsf(x);
            acc += fmaxf(x, 0.0f) + __logf(1.0f + __expf(-ax));
        }
    }

    // deterministic block tree reduction
    s_red[tid] = acc;
    __syncthreads();
    for (int s = 256; s > 0; s >>= 1) {
        if (tid < s) s_red[tid] += s_red[tid + s];
        __syncthreads();
    }
    if (tid == 0) {
        row_loss[m] = s_red[0];
        row_pair[m] = (float)npos * (float)nneg;    // exact (<= 65536)
    }
}

// ---------------------------------------------------------------------------
// Kernel 3: reduce 256 row partials, apply pair_num > 0 guard.
// ---------------------------------------------------------------------------
__global__ __launch_bounds__(256)
void finalize_kernel(const float* __restrict__ row_loss,
                     const float* __restrict__ row_pair,
                     float*       __restrict__ out)
{
    __shared__ float sl[256];
    __shared__ float sp[256];
    const int tid = threadIdx.x;
    sl[tid] = row_loss[tid];
    sp[tid] = row_pair[tid];
    __syncthreads();
    for (int s = 128; s > 0; s >>= 1) {
        if (tid < s) { sl[tid] += sl[tid + s]; sp[tid] += sp[tid + s]; }
        __syncthreads();
    }
    if (tid == 0)
        out[0] = (sp[0] > 0.0f) ? (sl[0] / sp[0]) : sl[0];
}

// ---------------------------------------------------------------------------
// Launch
// ---------------------------------------------------------------------------
extern "C" void kernel_launch(void* const* d_in, const int* in_sizes, int n_in,
                              void* d_out, int out_size, void* d_ws, size_t ws_size,
                              hipStream_t stream)
{
    const float* fo     = (const float*)d_in[0];   // features_o [B, D]
    const float* fi     = (const float*)d_in[1];   // features_i [B, D]
    const float* mask   = (const float*)d_in[2];   // mask       [B, B]
    const int*   msents = (const int*)  d_in[3];   // mask_sents [M]

    float* sim      = (float*)d_ws;                      // [M, B]
    float* row_loss = sim + (size_t)M_SZ * B_SZ;         // [M]
    float* row_pair = row_loss + M_SZ;                   // [M]

    dim3 g1(B_SZ / 16, M_SZ / 16);   // 32 x 16 tiles, one wave each
    sim_wmma_kernel<<<g1, 32, 0, stream>>>(fo, fi, msents, sim);
    row_loss_kernel<<<M_SZ, 512, 0, stream>>>(sim, mask, msents, row_loss, row_pair);
    finalize_kernel<<<1, 256, 0, stream>>>(row_loss, row_pair, (float*)d_out);
}